// TopK_63127429317014
// MI455X (gfx1250) — compile-verified
//
#include <hip/hip_runtime.h>
#include <stdint.h>

// Top-K (K=64) row masking: out[r, c] = x[r, c] if x[r, c] is among the 64
// largest in row r, else 0.  16384 rows x 4096 cols, fp32.
//
// Strategy (memory-bound, ~512 MiB moved -> ~23 us at 23.3 TB/s):
//  * one 256-thread workgroup per row
//  * stage the row into LDS with CDNA5 async global->LDS b128 copies
//    (ASYNCcnt / s_wait_asynccnt), no VGPR round-trip
//  * exact threshold via 4-pass 8-bit radix select using an LDS histogram
//  * masked, coalesced b128 stores

#define COLS    4096
#define KSEL    64
#define THREADS 256

__device__ __forceinline__ uint32_t fmap(uint32_t b) {
    // Order-preserving map: larger float <=> larger unsigned.
    return (b & 0x80000000u) ? ~b : (b | 0x80000000u);
}

__global__ __launch_bounds__(THREADS)
void TopK_63127429317014_kernel(const float* __restrict__ x,
                                float* __restrict__ out) {
    __shared__ uint32_t srow[COLS];      // raw fp32 bits of this row (16 KB)
    __shared__ uint32_t hist[256];       // radix histogram
    __shared__ uint32_t sh_prefix;
    __shared__ uint32_t sh_k;

    const int tid = threadIdx.x;
    const int row = blockIdx.x;
    const float* __restrict__ xr = x + (size_t)row * COLS;
    float* __restrict__ orow     = out + (size_t)row * COLS;

    // ---- 1. Async copy row -> LDS (4 x b128 per thread, perfectly coalesced)
#if defined(__gfx1250__)
    {
        #pragma unroll
        for (int j = 0; j < 4; ++j) {
            const int v = tid + j * THREADS;                 // float4 index [0,1024)
            const uint64_t ga = (uint64_t)(uintptr_t)(xr + (size_t)v * 4);
            const uint32_t la = (uint32_t)(uintptr_t)(&srow[v * 4]); // low 32b = LDS offset
            asm volatile("global_load_async_to_lds_b128 %0, %1, off"
                         :: "v"(la), "v"(ga) : "memory");
        }
        asm volatile("s_wait_asynccnt 0" ::: "memory");
    }
#else
    #pragma unroll
    for (int j = 0; j < 4; ++j) {
        const int v = tid + j * THREADS;
        const uint4 d = ((const uint4*)xr)[v];
        ((uint4*)srow)[v] = d;
    }
#endif
    __syncthreads();

    // ---- 2. Radix select: find u_thr = K-th largest mapped value
    uint32_t prefix = 0;
    uint32_t k      = KSEL;   // how many still needed at/below current prefix
    #pragma unroll
    for (int pass = 0; pass < 4; ++pass) {
        const int shift = 24 - 8 * pass;

        hist[tid] = 0;        // THREADS == 256 == #bins
        __syncthreads();

        for (int i = tid; i < COLS; i += THREADS) {
            const uint32_t u = fmap(srow[i]);
            const bool cand = (pass == 0) ||
                              ((u >> (shift + 8)) == (prefix >> (shift + 8)));
            if (cand) atomicAdd(&hist[(u >> shift) & 255u], 1u);
        }
        __syncthreads();

        if (tid == 0) {
            uint32_t cum = 0;      // elements strictly greater (higher bins)
            uint32_t b   = 0;
            for (int bin = 255; bin >= 0; --bin) {
                const uint32_t c = hist[bin];
                if (cum + c >= k) { b = (uint32_t)bin; k -= cum; break; }
                cum += c;
            }
            prefix |= b << shift;
            sh_prefix = prefix;
            sh_k      = k;
        }
        __syncthreads();
        prefix = sh_prefix;
        k      = sh_k;
        __syncthreads();
    }

    const uint32_t u_thr = prefix;  // exact bit pattern of K-th largest
    const uint32_t keq   = k;       // # of threshold-equal elements to keep

    // ---- 3. Masked output, b128 stores.  Ties kept by lowest index
    //         (matches lax.top_k stable ordering); ties are rare so the
    //         rank scan below is effectively never taken.
    #pragma unroll
    for (int j = 0; j < 4; ++j) {
        const int v = tid + j * THREADS;    // float4 index
        float4 o;
        float* op = (float*)&o;
        #pragma unroll
        for (int c = 0; c < 4; ++c) {
            const int i = v * 4 + c;
            const uint32_t b = srow[i];
            const uint32_t u = fmap(b);
            float val = 0.0f;
            if (u > u_thr) {
                val = __uint_as_float(b);
            } else if (u == u_thr) {
                uint32_t r = 0;
                for (int jj = 0; jj < i; ++jj)
                    if (fmap(srow[jj]) == u_thr) ++r;
                if (r < keq) val = __uint_as_float(b);
            }
            op[c] = val;
        }
        ((float4*)orow)[v] = o;
    }
}

extern "C" void kernel_launch(void* const* d_in, const int* in_sizes, int n_in,
                              void* d_out, int out_size, void* d_ws, size_t ws_size,
                              hipStream_t stream) {
    (void)in_sizes; (void)n_in; (void)d_ws; (void)ws_size;
    const float* x = (const float*)d_in[0];
    float* out     = (float*)d_out;
    const int rows = out_size / COLS;     // 16384
    TopK_63127429317014_kernel<<<rows, THREADS, 0, stream>>>(x, out);
}